// GriffinGate_23012434772060
// MI455X (gfx1250) — compile-verified
//
#include <hip/hip_runtime.h>
#include <hip/hip_bf16.h>

// ---------------------------------------------------------------------------
// GriffinGate fused kernel for gfx1250 (MI455X, wave32, WMMA).
//
// Pass 1: per-row LayerNorm statistics (mean, rstd) -> d_ws
// Pass 2: fused LN-apply + block-diagonal matmul (V_WMMA_F32_16X16X4_F32)
//         + gate chain, writing alpha and beta with NT stores (outputs are
//         never re-read; keep the 192MB L2 for x / w / ln tables instead of
//         letting 256MB of output sweep it).
//
// Shapes: x (4,8192,1024) fp32 -> 32768 rows of width 1024.
// Block-diag: 16 blocks of 64x64. One wave handles (16-row group, block):
// a 16x64 output tile = 4 WMMA accumulators, K=64 chained as 16 wmma steps.
// ---------------------------------------------------------------------------

typedef __attribute__((ext_vector_type(2))) float v2f;
typedef __attribute__((ext_vector_type(8))) float v8f;

#define ROWS      32768
#define WIDTH     1024
#define NBLK      16
#define BLKW      64
#define C_VALUE   8.0f
#define LN_EPS    1e-6f

// ---------------- Pass 1: LayerNorm stats (one wave per row) ----------------
__global__ __launch_bounds__(256) void ln_stats_kernel(
    const float* __restrict__ x, float2* __restrict__ stats)
{
    const int lane = threadIdx.x & 31;
    const int row  = blockIdx.x * 8 + (threadIdx.x >> 5);   // 8 waves / block
    const float* xr = x + (size_t)row * WIDTH;

    float s = 0.0f, sq = 0.0f;
#pragma unroll
    for (int i = 0; i < 8; ++i) {
        // coalesced: 32 lanes * 16B = 512B per iteration
        float4 v = *(const float4*)(xr + lane * 4 + i * 128);
        s  += v.x + v.y + v.z + v.w;
        sq += v.x * v.x + v.y * v.y + v.z * v.z + v.w * v.w;
    }
#pragma unroll
    for (int off = 16; off >= 1; off >>= 1) {
        s  += __shfl_xor(s,  off, 32);
        sq += __shfl_xor(sq, off, 32);
    }
    if (lane == 0) {
        const float mean = s * (1.0f / WIDTH);
        const float var  = sq * (1.0f / WIDTH) - mean * mean;
        stats[row] = make_float2(mean, rsqrtf(var + LN_EPS));
    }
}

// ---------------- Pass 2 epilogue: gate + NT store ----------------
__device__ __forceinline__ void gate_store(
    v8f acc, int col, int rowg, int half,
    const float* __restrict__ bias, const float* __restrict__ a_param,
    float* __restrict__ out)
{
    const size_t TOT = (size_t)ROWS * WIDTH;  // alpha size; beta follows
    const float bv = bias[col];
    const float ap = a_param[col];
    const float sp = __logf(1.0f + __expf(ap));       // softplus(a_param)
    float* o = out + (size_t)(rowg + half * 8) * WIDTH + col;
#pragma unroll
    for (int v = 0; v < 8; ++v) {                     // M = half*8 + v
        const float y     = acc[v] + bv;
        const float sig   = 1.0f / (1.0f + __expf(-y));
        float alpha       = __expf(-C_VALUE * sig * sp);
        alpha             = fminf(fmaxf(alpha, 0.05f), 0.95f);
        // Outputs are write-once: non-temporal so they don't evict x/w in L2.
        __builtin_nontemporal_store(alpha,        o + (size_t)v * WIDTH);
        __builtin_nontemporal_store(1.0f - alpha, o + TOT + (size_t)v * WIDTH);
    }
}

// ---------------- Pass 2: fused LN + block-diag WMMA + gate ----------------
__global__ __launch_bounds__(256) void griffin_wmma_kernel(
    const float* __restrict__ x,
    const float* __restrict__ ln_scale,
    const float* __restrict__ ln_bias,
    const float* __restrict__ w,
    const float* __restrict__ b,
    const float* __restrict__ a_param,
    const float2* __restrict__ stats,
    float* __restrict__ out)
{
    const int lane  = threadIdx.x & 31;
    const int gwave = blockIdx.x * 8 + (threadIdx.x >> 5);  // 0..32767
    const int g     = gwave >> 4;            // 16-row group, 0..2047
    const int h     = gwave & 15;            // block index,  0..15

    const int half = lane >> 4;              // 0 or 1 (upper half-wave)
    const int lid  = lane & 15;              // M index (A) / N index (B,C,D)
    const int kk   = half * 2;               // K sub-offset within step of 4

    const int rowg = g * 16;

    // Per-lane row stats: A-fragment row M == lid for both half-waves.
    const float2 st  = stats[rowg + lid];
    const float mean = st.x, rstd = st.y;

    const float* xrow = x + (size_t)(rowg + lid) * WIDTH + h * BLKW;
    const float* wblk = w + h * (BLKW * BLKW);

    v8f acc0 = {}, acc1 = {}, acc2 = {}, acc3 = {};

#pragma unroll 4
    for (int ks = 0; ks < 16; ++ks) {        // K = 64 = 16 steps of 4
        const int k0 = ks * 4;
        const int c  = h * BLKW + k0 + kk;   // global column of a.x

        // A fragment (16x4 f32): lane -> row M=lid, cols (k0+kk, k0+kk+1)
        const float2 xv = *(const float2*)(xrow + k0 + kk);
        const float2 ls = *(const float2*)(ln_scale + c);
        const float2 lb = *(const float2*)(ln_bias + c);
        v2f a;
        a.x = (xv.x - mean) * rstd * ls.x + lb.x;
        a.y = (xv.y - mean) * rstd * ls.y + lb.y;

        // B fragments (4x16 f32) for the four N-subtiles of this block.
        // b.x = w[h][k0+kk][nt*16+lid], b.y = w[h][k0+kk+1][nt*16+lid]
        const float* wrow = wblk + (k0 + kk) * BLKW + lid;
        v2f b0, b1, b2, b3;
        b0.x = wrow[0];   b0.y = wrow[64];
        b1.x = wrow[16];  b1.y = wrow[80];
        b2.x = wrow[32];  b2.y = wrow[96];
        b3.x = wrow[48];  b3.y = wrow[112];

        acc0 = __builtin_amdgcn_wmma_f32_16x16x4_f32(false, a, false, b0,
                                                     (short)0, acc0, false, false);
        acc1 = __builtin_amdgcn_wmma_f32_16x16x4_f32(false, a, false, b1,
                                                     (short)0, acc1, false, false);
        acc2 = __builtin_amdgcn_wmma_f32_16x16x4_f32(false, a, false, b2,
                                                     (short)0, acc2, false, false);
        acc3 = __builtin_amdgcn_wmma_f32_16x16x4_f32(false, a, false, b3,
                                                     (short)0, acc3, false, false);
    }

    const int colBase = h * BLKW + lid;
    gate_store(acc0, colBase +  0, rowg, half, b, a_param, out);
    gate_store(acc1, colBase + 16, rowg, half, b, a_param, out);
    gate_store(acc2, colBase + 32, rowg, half, b, a_param, out);
    gate_store(acc3, colBase + 48, rowg, half, b, a_param, out);
}

// ---------------------------------------------------------------------------
extern "C" void kernel_launch(void* const* d_in, const int* in_sizes, int n_in,
                              void* d_out, int out_size, void* d_ws, size_t ws_size,
                              hipStream_t stream)
{
    const float* x        = (const float*)d_in[0];  // (4,8192,1024)
    const float* ln_scale = (const float*)d_in[1];  // (1024,)
    const float* ln_bias  = (const float*)d_in[2];  // (1024,)
    const float* w        = (const float*)d_in[3];  // (16,64,64)
    const float* b        = (const float*)d_in[4];  // (16,64)
    const float* a_param  = (const float*)d_in[5];  // (1024,)
    float* out            = (float*)d_out;          // alpha | beta, 2*ROWS*WIDTH
    float2* stats         = (float2*)d_ws;          // ROWS float2 = 256 KB

    // Pass 1: 32768 rows, 8 rows per 256-thread block.
    ln_stats_kernel<<<ROWS / 8, 256, 0, stream>>>(x, stats);

    // Pass 2: 32768 waves (2048 row-groups x 16 blocks), 8 waves per block.
    griffin_wmma_kernel<<<ROWS / 8, 256, 0, stream>>>(
        x, ln_scale, ln_bias, w, b, a_param, stats, out);
}